// GATLayer_48387101556917
// MI455X (gfx1250) — compile-verified
//
#include <hip/hip_runtime.h>

#define NNODES 100000
#define NEDGES 1600000
#define CIN    256
#define NHEADS 2
#define DHEAD  128
#define FEAT   (NHEADS * DHEAD)   // 256
#define ALPHA  0.2f

typedef float v2f __attribute__((ext_vector_type(2)));
typedef float v8f __attribute__((ext_vector_type(8)));

// ---------------- Kernel 0: zero the accumulators (d_out reused as `num`) ----
__global__ __launch_bounds__(256) void gat_zero_kernel(float* __restrict__ out_acc,
                                                       float* __restrict__ rowsum) {
    int tid = blockIdx.x * blockDim.x + threadIdx.x;
    if (tid < NNODES * FEAT) out_acc[tid] = 0.0f;
    if (tid < NNODES * NHEADS) rowsum[tid] = 0.0f;
}

// ---------------- Kernel 1: projection GEMM (fp32 WMMA) + attention scores ---
// grid = (ceil(N/128), NHEADS), block = 256 threads = 8 wave32.
// Wave w computes rows [strip*128 + w*16, +16) x all 128 output cols of one head.
#define KCHUNK     64
#define LDS_STRIDE 68   // 64 + 4 pad: banks 4 apart across cols; 68*4=272 (16B multiple)

__global__ __launch_bounds__(256) void gat_gemm_kernel(
    const float* __restrict__ x,   // [N, CIN]
    const float* __restrict__ Wg,  // [H, DHEAD, CIN]
    const float* __restrict__ bg,  // [H, DHEAD]
    const float* __restrict__ ag,  // [H, 2*DHEAD]
    float* __restrict__ hbuf,      // [N, H*DHEAD]
    float* __restrict__ s_src,     // [N, H]
    float* __restrict__ s_dst)     // [N, H]
{
    __shared__ float Wlds[DHEAD * LDS_STRIDE];   // 34,816 bytes

    const int head  = blockIdx.y;
    const int strip = blockIdx.x;
    const int tid   = threadIdx.x;
    const int wave  = tid >> 5;
    const int lane  = tid & 31;
    const int lhalf = lane & 15;              // n (B/C col) or m (A row) within tile
    const int kOff  = (lane >> 4) << 1;       // lanes 0-15: K+{0,1}; lanes 16-31: K+{2,3}
    const int rowA  = min(strip * 128 + wave * 16 + lhalf, NNODES - 1);

    v8f acc[8] = {};                          // 8 col-tiles of 16x16 fp32 C

    for (int kc = 0; kc < CIN / KCHUNK; ++kc) {
        const int kbase = kc * KCHUNK;
        __syncthreads();
        // Cooperative stage of W[head][0..127][kbase..kbase+63] into LDS.
        #pragma unroll
        for (int j = 0; j < 8; ++j) {
            int linear = j * 256 + tid;          // 0..2047, 4 floats each
            int col    = linear >> 4;            // 0..127
            int kk4    = (linear & 15) << 2;     // 0,4,...,60
            const float4 w4 = *reinterpret_cast<const float4*>(
                &Wg[(size_t)(head * DHEAD + col) * CIN + kbase + kk4]);
            *reinterpret_cast<float4*>(&Wlds[col * LDS_STRIDE + kk4]) = w4;
        }
        __syncthreads();

        for (int kt = 0; kt < KCHUNK / 4; ++kt) {
            const int kk = kt * 4 + kOff;
            // A-frag: 16x4 fp32, lane layout per ISA (M = lane%16, K split by half-wave)
            const v2f afrag = *reinterpret_cast<const v2f*>(
                &x[(size_t)rowA * CIN + kbase + kk]);
            #pragma unroll
            for (int t = 0; t < 8; ++t) {
                // B-frag: 4x16 fp32, B[k][n] = W[head][colBase+n][k]
                const v2f bfrag = *reinterpret_cast<const v2f*>(
                    &Wlds[(t * 16 + lhalf) * LDS_STRIDE + kk]);
                acc[t] = __builtin_amdgcn_wmma_f32_16x16x4_f32(
                    false, afrag, false, bfrag, (short)0, acc[t], false, false);
            }
        }
    }

    // C layout: acc[t][i] = h[rowBase + i][t*16 + lhalf], rowBase has +8 for upper half.
    const int rowBase = strip * 128 + wave * 16 + ((lane >> 4) << 3);

    float psrc[8] = {}, pdst[8] = {};
    #pragma unroll
    for (int t = 0; t < 8; ++t) {
        const int col    = t * 16 + lhalf;
        const float bias = bg[head * DHEAD + col];
        const float asrc = ag[head * 2 * DHEAD + col];
        const float adst = ag[head * 2 * DHEAD + DHEAD + col];
        #pragma unroll
        for (int i = 0; i < 8; ++i) {
            const float hv = acc[t][i] + bias;
            const int row  = rowBase + i;
            if (row < NNODES)
                hbuf[(size_t)row * FEAT + head * DHEAD + col] = hv;
            psrc[i] += hv * asrc;
            pdst[i] += hv * adst;
        }
    }
    // Reduce the col dimension across the 16 lanes of each half-wave.
    #pragma unroll
    for (int off = 1; off < 16; off <<= 1) {
        #pragma unroll
        for (int i = 0; i < 8; ++i) {
            psrc[i] += __shfl_xor(psrc[i], off, 16);
            pdst[i] += __shfl_xor(pdst[i], off, 16);
        }
    }
    if (lhalf == 0) {
        #pragma unroll
        for (int i = 0; i < 8; ++i) {
            const int row = rowBase + i;
            if (row < NNODES) {
                s_src[row * NHEADS + head] = psrc[i];
                s_dst[row * NHEADS + head] = pdst[i];
            }
        }
    }
}

// ---------------- Kernel 2: per-edge attention + scatter-add aggregation -----
// One wave32 per edge. h (102MB) is L2-resident; atomics resolve in L2.
__global__ __launch_bounds__(256) void gat_edge_kernel(
    const int* __restrict__ ei,       // [2, E]
    const float* __restrict__ hbuf,   // [N, H*DHEAD]
    const float* __restrict__ s_src,  // [N, H]
    const float* __restrict__ s_dst,  // [N, H]
    float* __restrict__ rowsum,       // [N, H]
    float* __restrict__ out_acc)      // [N, H*DHEAD] (num accumulator)
{
    const int e = blockIdx.x * 8 + (threadIdx.x >> 5);
    if (e >= NEDGES) return;
    const int lane = threadIdx.x & 31;
    const int src  = ei[e];
    const int dst  = ei[NEDGES + e];

    #pragma unroll
    for (int head = 0; head < NHEADS; ++head) {
        const float sc = s_src[src * NHEADS + head] + s_dst[dst * NHEADS + head];
        const float lr = sc > 0.0f ? sc : ALPHA * sc;
        const float w  = expf(-lr);
        if (lane == 0) atomicAdd(&rowsum[src * NHEADS + head], w);
        const float4 hv = *reinterpret_cast<const float4*>(
            &hbuf[(size_t)dst * FEAT + head * DHEAD + lane * 4]);
        float* dp = &out_acc[(size_t)src * FEAT + head * DHEAD + lane * 4];
        atomicAdd(dp + 0, w * hv.x);
        atomicAdd(dp + 1, w * hv.y);
        atomicAdd(dp + 2, w * hv.z);
        atomicAdd(dp + 3, w * hv.w);
    }
}

// ---------------- Kernel 3: normalize + ELU in place -------------------------
__global__ __launch_bounds__(256) void gat_final_kernel(float* __restrict__ out,
                                                        const float* __restrict__ rowsum) {
    const int tid = blockIdx.x * blockDim.x + threadIdx.x;
    if (tid >= NNODES * FEAT) return;
    const int node = tid >> 8;          // / 256
    const int head = (tid >> 7) & 1;    // (tid % 256) / 128
    const float v = out[tid] / rowsum[node * NHEADS + head];
    out[tid] = v > 0.0f ? v : (expf(v) - 1.0f);   // jax.nn.elu, alpha = 1
}

// ---------------- Launcher ---------------------------------------------------
extern "C" void kernel_launch(void* const* d_in, const int* in_sizes, int n_in,
                              void* d_out, int out_size, void* d_ws, size_t ws_size,
                              hipStream_t stream) {
    const float* x  = (const float*)d_in[0];
    const int*   ei = (const int*)d_in[1];
    const float* Wg = (const float*)d_in[2];
    const float* bg = (const float*)d_in[3];
    const float* ag = (const float*)d_in[4];
    float* out = (float*)d_out;

    char* ws = (char*)d_ws;
    float* hbuf   = (float*)ws;                                          // N*256 f32
    float* s_srcb = (float*)(ws + (size_t)NNODES * FEAT * sizeof(float));
    float* s_dstb = s_srcb + NNODES * NHEADS;
    float* rowsum = s_dstb + NNODES * NHEADS;

    gat_zero_kernel<<<(NNODES * FEAT + 255) / 256, 256, 0, stream>>>(out, rowsum);

    dim3 g1((NNODES + 127) / 128, NHEADS);
    gat_gemm_kernel<<<g1, 256, 0, stream>>>(x, Wg, bg, ag, hbuf, s_srcb, s_dstb);

    gat_edge_kernel<<<(NEDGES + 7) / 8, 256, 0, stream>>>(ei, hbuf, s_srcb, s_dstb,
                                                          rowsum, out);

    gat_final_kernel<<<(NNODES * FEAT + 255) / 256, 256, 0, stream>>>(out, rowsum);
}